// HetGCN_50843822850190
// MI455X (gfx1250) — compile-verified
//
#include <hip/hip_runtime.h>

#define DD 128   // feature dim

typedef __attribute__((ext_vector_type(2))) float v2f;
typedef __attribute__((ext_vector_type(4))) float v4f;
typedef __attribute__((ext_vector_type(8))) float v8f;

// ---------------------------------------------------------------- utilities

__global__ void zero_f32(float* __restrict__ p, long n) {
    long i = (long)blockIdx.x * blockDim.x + threadIdx.x;
    long stride = (long)gridDim.x * blockDim.x;
    for (; i < n; i += stride) p[i] = 0.0f;
}

// ------------------------------------------------- content GEMM (WMMA f32)
// H[n][o] = LeakyReLU( sum_d X[n][d] * W[o][d] + bias[o] ), W row-major [o][d]
// One block = 256 threads = 8 waves; block covers 128 rows, wave w owns rows
// [blk*128 + w*16, +16). B panel (128x128, already [col][k]) staged in LDS.
__global__ __launch_bounds__(256)
void content_gemm(const float* __restrict__ X, const float* __restrict__ W,
                  const float* __restrict__ bias, float* __restrict__ H, int N) {
    __shared__ float Bt[DD * DD];              // Bt[o*128 + k] = W[o][k]
    {
        const v4f* Wv = (const v4f*)W;
        v4f* Bv = (v4f*)Bt;
#pragma unroll
        for (int i = 0; i < (DD * DD / 4) / 256; ++i)
            Bv[threadIdx.x + i * 256] = Wv[threadIdx.x + i * 256];
    }
    __syncthreads();

    const int lane = threadIdx.x & 31;
    const int wv   = threadIdx.x >> 5;
    const int lh   = lane >> 4;                // lane half: 0 or 1
    const int l16  = lane & 15;
    const int rowt = blockIdx.x * 128 + wv * 16;   // 16-row tile base
    const int arow = rowt + l16;               // this lane's A row (whole tile valid or not)
    const bool valid = (arow < N);
    const float* xrow = X + (long)arow * DD;

    // Preload all A fragments for this row tile: a2[kc] covers K = kc*4 + 2*lh + {0,1}
    v2f a2[32];
#pragma unroll
    for (int kc = 0; kc < 32; ++kc) {
        v2f a = {0.0f, 0.0f};
        if (valid) a = *(const v2f*)(xrow + kc * 4 + 2 * lh);
        a2[kc] = a;
    }

#pragma unroll
    for (int ct = 0; ct < 8; ++ct) {
        const int bn = ct * 16 + l16;          // output column for this lane
        v8f acc = {};
#pragma unroll
        for (int kc = 0; kc < 32; ++kc) {
            v2f b = *(const v2f*)&Bt[bn * DD + kc * 4 + 2 * lh];
            acc = __builtin_amdgcn_wmma_f32_16x16x4_f32(
                false, a2[kc], false, b, (short)0, acc, false, false);
        }
        const float bcol = bias[bn];
#pragma unroll
        for (int v = 0; v < 8; ++v) {
            const int m = v + 8 * lh;          // row within tile
            float y = acc[v] + bcol;
            y = (y > 0.0f) ? y : 0.01f * y;    // leaky_relu, slope 0.01
            H[(long)(rowt + m) * DD + bn] = y; // H is padded; store unguarded
        }
    }
}

// --------------------------------------------- accumulate GEMM (WMMA f32)
// Z[n][o] += sum_k A[n][k] * Wagg[(row0+k)*128 + o]   (A, Z are NPAD-padded)
__global__ __launch_bounds__(256)
void agg_gemm(const float* __restrict__ A, const float* __restrict__ Wagg,
              int row0, float* __restrict__ Z) {
    __shared__ float Bt[DD * DD];              // Bt[o*128 + k] = Wagg[(row0+k)*128 + o]
#pragma unroll
    for (int i = 0; i < (DD * DD) / 256; ++i) {
        const int idx = threadIdx.x + i * 256;
        const int o = idx & 127, k = idx >> 7; // coalesced global read over o
        Bt[o * DD + k] = Wagg[(long)(row0 + k) * DD + o];
    }
    __syncthreads();

    const int lane = threadIdx.x & 31;
    const int wv   = threadIdx.x >> 5;
    const int lh   = lane >> 4;
    const int l16  = lane & 15;
    const int rowt = blockIdx.x * 128 + wv * 16;
    const float* arow = A + (long)(rowt + l16) * DD;

    v2f a2[32];
#pragma unroll
    for (int kc = 0; kc < 32; ++kc)
        a2[kc] = *(const v2f*)(arow + kc * 4 + 2 * lh);

#pragma unroll
    for (int ct = 0; ct < 8; ++ct) {
        const int bn = ct * 16 + l16;
        v8f acc;
#pragma unroll
        for (int v = 0; v < 8; ++v)            // C = current Z tile (accumulate)
            acc[v] = Z[(long)(rowt + v + 8 * lh) * DD + bn];
#pragma unroll
        for (int kc = 0; kc < 32; ++kc) {
            v2f b = *(const v2f*)&Bt[bn * DD + kc * 4 + 2 * lh];
            acc = __builtin_amdgcn_wmma_f32_16x16x4_f32(
                false, a2[kc], false, b, (short)0, acc, false, false);
        }
#pragma unroll
        for (int v = 0; v < 8; ++v)
            Z[(long)(rowt + v + 8 * lh) * DD + bn] = acc[v];
    }
}

// ------------------------------------------------------- scatter (segment sum)
// One wave per edge: gather 512B row of H[src], atomic-add into slot[dst].
__global__ __launch_bounds__(256)
void scatter_edges(const float* __restrict__ H, const int* __restrict__ src,
                   const int* __restrict__ dst, float* __restrict__ slot,
                   float* __restrict__ cnt, int E) {
    const int lane = threadIdx.x & 31;
    const int e = (int)(((long)blockIdx.x * blockDim.x + threadIdx.x) >> 5);
    if (e >= E) return;
    const int s = src[e], d = dst[e];
    v4f v = *(const v4f*)(H + (long)s * DD + lane * 4);
    float* out = slot + (long)d * DD + lane * 4;
    unsafeAtomicAdd(out + 0, v[0]);
    unsafeAtomicAdd(out + 1, v[1]);
    unsafeAtomicAdd(out + 2, v[2]);
    unsafeAtomicAdd(out + 3, v[3]);
    if (lane == 0) unsafeAtomicAdd(cnt + d, 1.0f);
}

// slot[n][c] /= max(cnt[n], 1)
__global__ void divide_mean(float* __restrict__ slot, const float* __restrict__ cnt,
                            long total) {
    long i = (long)blockIdx.x * blockDim.x + threadIdx.x;
    long stride = (long)gridDim.x * blockDim.x;
    for (; i < total; i += stride)
        slot[i] /= fmaxf(cnt[i >> 7], 1.0f);
}

// out[n][c] = (n < N && node_type[n] == t) ? H[n][c] : 0
__global__ void mask_self(const float* __restrict__ H, const int* __restrict__ ntype,
                          float* __restrict__ out, int t, int N, long total) {
    long i = (long)blockIdx.x * blockDim.x + threadIdx.x;
    long stride = (long)gridDim.x * blockDim.x;
    for (; i < total; i += stride) {
        long n = i >> 7;
        out[i] = (n < N && ntype[n] == t) ? H[i] : 0.0f;
    }
}

// out[c] += (1/N) * sum_n sigmoid(Z[n][c] + b[c])   (out pre-zeroed)
__global__ __launch_bounds__(128)
void reduce_mean(const float* __restrict__ Z, const float* __restrict__ bagg,
                 float* __restrict__ out, int N) {
    const int c = threadIdx.x;
    const float b = bagg[c];
    float sum = 0.0f;
    for (int n = blockIdx.x; n < N; n += gridDim.x) {
        const float x = Z[(long)n * DD + c] + b;
        sum += 1.0f / (1.0f + expf(-x));
    }
    unsafeAtomicAdd(&out[c], sum * (1.0f / (float)N));
}

// ---------------------------------------------------------------- launcher

extern "C" void kernel_launch(void* const* d_in, const int* in_sizes, int n_in,
                              void* d_out, int out_size, void* d_ws, size_t ws_size,
                              hipStream_t stream) {
    const float* X     = (const float*)d_in[0];   // [N,128]
    const float* Wc    = (const float*)d_in[1];   // [T,128,128]
    const float* bc    = (const float*)d_in[2];   // [T,128]
    const float* Wagg  = (const float*)d_in[3];   // [640,128]
    const float* bagg  = (const float*)d_in[4];   // [128]
    const int*   esrc  = (const int*)d_in[5];     // [T,E]
    const int*   edst  = (const int*)d_in[6];     // [T,E]
    const int*   ntype = (const int*)d_in[7];     // [N]

    const int N = in_sizes[0] / DD;
    const int T = in_sizes[1] / (DD * DD);
    const int E = in_sizes[5] / T;
    const long NPAD = (((long)N + 127) / 128) * 128;

    float* H    = (float*)d_ws;                  // [NPAD,128] per-type content embed
    float* slot = H + NPAD * DD;                 // [NPAD,128] segment-mean / masked self
    float* Z    = slot + NPAD * DD;              // [NPAD,128] agg pre-activation
    float* cnt  = Z + NPAD * DD;                 // [NPAD]     edge counts

    const int gemmBlocks = (int)(NPAD / 128);
    const long elems = NPAD * DD;

    zero_f32<<<512, 256, 0, stream>>>(Z, elems);
    zero_f32<<<1, 128, 0, stream>>>((float*)d_out, out_size);

    for (int t = 0; t < T; ++t) {
        // H_t = leaky_relu(X @ W_t^T + b_t)
        content_gemm<<<gemmBlocks, 256, 0, stream>>>(
            X, Wc + (long)t * DD * DD, bc + (long)t * DD, H, N);
        // segment mean into slot
        zero_f32<<<512, 256, 0, stream>>>(slot, elems);
        zero_f32<<<64, 256, 0, stream>>>(cnt, NPAD);
        scatter_edges<<<(E + 7) / 8, 256, 0, stream>>>(
            H, esrc + (long)t * E, edst + (long)t * E, slot, cnt, E);
        divide_mean<<<1024, 256, 0, stream>>>(slot, cnt, elems);
        // Z += slot_t @ W_agg[t*128 : (t+1)*128, :]
        agg_gemm<<<gemmBlocks, 256, 0, stream>>>(slot, Wagg, t * DD, Z);
        // Z += (self where node_type==t) @ W_agg[T*128 : , :]
        mask_self<<<1024, 256, 0, stream>>>(H, ntype, slot, t, N, elems);
        agg_gemm<<<gemmBlocks, 256, 0, stream>>>(slot, Wagg, T * DD, Z);
    }

    reduce_mean<<<256, 128, 0, stream>>>(Z, bagg, (float*)d_out, N);
}